// LocalSelfAttention_81905026335368
// MI455X (gfx1250) — compile-verified
//
#include <hip/hip_runtime.h>
#include <math.h>

typedef __attribute__((ext_vector_type(2))) float v2f;
typedef __attribute__((ext_vector_type(8))) float v8f;

#define S_LEN 4096
#define BSZ   2
#define EMB   1024
#define HEADS 8
#define DH    128
#define WIN   256

__device__ __forceinline__ v8f wmma4f32(v2f a, v2f b, v8f c) {
  // V_WMMA_F32_16X16X4_F32: D = A(16x4) * B(4x16) + C(16x16), wave32
  return __builtin_amdgcn_wmma_f32_16x16x4_f32(false, a, false, b, (short)0, c,
                                               false, false);
}

// ---- CDNA5 async global->LDS copy (ASYNCcnt) with sync fallback -----------
#if __has_builtin(__builtin_amdgcn_global_load_async_to_lds_b128)
#define HAVE_ASYNC_LDS 1
typedef int vi4 __attribute__((vector_size(16)));
typedef vi4 __attribute__((address_space(1))) *gv4p;   // global int4*
typedef vi4 __attribute__((address_space(3))) *lv4p;   // LDS int4*
__device__ __forceinline__ void cp16_g2l(const float* g, float* l) {
  __builtin_amdgcn_global_load_async_to_lds_b128((gv4p)(void*)g, (lv4p)(void*)l,
                                                 0, 0);
}
__device__ __forceinline__ void async_wait0() {
#if __has_builtin(__builtin_amdgcn_s_wait_asynccnt)
  __builtin_amdgcn_s_wait_asynccnt(0);
#else
  asm volatile("s_wait_asynccnt 0x0" ::: "memory");
#endif
}
#else
#define HAVE_ASYNC_LDS 0
__device__ __forceinline__ void cp16_g2l(const float* g, float* l) {
  float4 t = *(const float4*)g;
  *(float4*)l = t;
}
__device__ __forceinline__ void async_wait0() {}
#endif

// ---------------------------------------------------------------------------
// Kernel 1: fused QKV projection GEMM, Y = permute_rows(X) @ W^T + bias
// Output layout: buf[(b'*H + h)][s'][d].  The torch ".view scramble" is folded
// into the row map: row m = b'*4096 + s' -> val row b = s'/2048,
// s = 2*(s'%2048) + b'.  Double-buffered async LDS staging, 1 barrier/iter.
// ---------------------------------------------------------------------------
__global__ __launch_bounds__(128, 1)
void qkv_proj_kernel(const float* __restrict__ val,
                     const float* __restrict__ Wq, const float* __restrict__ bq,
                     const float* __restrict__ Wk, const float* __restrict__ bk,
                     const float* __restrict__ Wv, const float* __restrict__ bv,
                     float* __restrict__ qbuf, float* __restrict__ kbuf,
                     float* __restrict__ vbuf)
{
  __shared__ float Xs[2][64 * 20];   // stride 20: conflict-free, 16B-aligned rows
  __shared__ float Ws[2][64 * 20];

  const int z = blockIdx.z;
  const float* W    = (z == 0) ? Wq : (z == 1) ? Wk : Wv;
  const float* bias = (z == 0) ? bq : (z == 1) ? bk : bv;
  float* outb       = (z == 0) ? qbuf : (z == 1) ? kbuf : vbuf;
  const float scale = (z == 0) ? 0.08838834764831845f : 1.0f;  // 1/sqrt(128)

  const int m0 = blockIdx.x * 64;   // row block (of 8192)
  const int n0 = blockIdx.y * 64;   // col block (of 1024)

  const int tid  = threadIdx.x;
  const int lane = tid & 31;
  const int wave = tid >> 5;        // 0..3: each wave 16 rows x 64 cols
  const int ln   = lane & 15;
  const int hi   = lane >> 4;

  // loader mapping: thread -> (row, 8-float column chunk)
  const int lrow = tid >> 1;         // 0..63
  const int lcol = (tid & 1) * 8;    // 0 or 8
  const int gm   = m0 + lrow;
  const int bp   = gm >> 12;         // b'
  const int sp   = gm & 4095;        // s'
  const int bb   = sp >> 11;         // source batch
  const int ss   = ((sp & 2047) << 1) + bp;  // source seq pos
  const float* xrow = val + ((size_t)ss * BSZ + bb) * EMB + lcol;
  const float* wrow = W + (size_t)(n0 + lrow) * EMB + lcol;
  const int ldst = lrow * 20 + lcol;

  v8f acc[4];
#pragma unroll
  for (int t = 0; t < 4; ++t)
#pragma unroll
    for (int r = 0; r < 8; ++r) acc[t][r] = 0.0f;

  const int am = wave * 16 + ln;

  // prologue: stage K-chunk 0 into buffer 0
  cp16_g2l(xrow, &Xs[0][ldst]);
  cp16_g2l(xrow + 4, &Xs[0][ldst + 4]);
  cp16_g2l(wrow, &Ws[0][ldst]);
  cp16_g2l(wrow + 4, &Ws[0][ldst + 4]);

  for (int st = 0; st < 64; ++st) {
    async_wait0();        // my async copies for chunk st have landed
    __syncthreads();      // everyone's landed; prev reads of other buffer done

    if (st + 1 < 64) {    // issue chunk st+1 into the other buffer
      const int kk = (st + 1) * 16;
      const int nb = (st + 1) & 1;
      cp16_g2l(xrow + kk, &Xs[nb][ldst]);
      cp16_g2l(xrow + kk + 4, &Xs[nb][ldst + 4]);
      cp16_g2l(wrow + kk, &Ws[nb][ldst]);
      cp16_g2l(wrow + kk + 4, &Ws[nb][ldst + 4]);
    }

    const float* xs = Xs[st & 1];
    const float* ws = Ws[st & 1];
#pragma unroll
    for (int j = 0; j < 4; ++j) {
      v2f a = *(const v2f*)&xs[am * 20 + j * 4 + 2 * hi];
#pragma unroll
      for (int t = 0; t < 4; ++t) {
        v2f b = *(const v2f*)&ws[(t * 16 + ln) * 20 + j * 4 + 2 * hi];
        acc[t] = wmma4f32(a, b, acc[t]);
      }
    }
  }

  // epilogue: bias, scale, store into [(b'*H+h)][s'][d]
#pragma unroll
  for (int t = 0; t < 4; ++t) {
    const int n = n0 + t * 16 + ln;
    const float bval = bias[n];
    const int h = n >> 7;
    const int d = n & 127;
#pragma unroll
    for (int r = 0; r < 8; ++r) {
      const int m   = m0 + wave * 16 + r + 8 * hi;
      const int obp = m >> 12;
      const int osp = m & 4095;
      outb[(((size_t)obp * HEADS + h) * S_LEN + osp) * DH + d] =
          (acc[t][r] + bval) * scale;
    }
  }
}

// ---------------------------------------------------------------------------
// Kernel 2: banded attention with online (flash-style) softmax.
// One WG = 8 waves = 128 consecutive queries of one (b',h).
// 40 absolute 16-key chunks; K/V chunk double-buffered in LDS via async
// copies (zero ds_store for OOB keys -> score 0 enters softmax, matching the
// reference's zero-padding); out-of-band strip columns masked to -1e30.
// ---------------------------------------------------------------------------
__global__ __launch_bounds__(256, 1)
void banded_attn_kernel(const float* __restrict__ qbuf,
                        const float* __restrict__ kbuf,
                        const float* __restrict__ vbuf,
                        float* __restrict__ out)
{
  __shared__ float Ks[2][16 * 132];    // stride 132: conflict-free B-frag reads
  __shared__ float Vs[2][16 * 132];
  __shared__ float Ps[8][16 * 20];     // per-wave P tile bounce (D->A layout)

  const int bh = blockIdx.y;           // b'*8 + h
  const int qb = blockIdx.x * 128;     // query base of this WG
  const int tid  = threadIdx.x;
  const int lane = tid & 31;
  const int wave = tid >> 5;           // 0..7
  const int ln = lane & 15;
  const int hi = lane >> 4;
  const int s0 = qb + wave * 16;       // this wave's 16-query tile

  const float* Qb = qbuf + ((size_t)bh * S_LEN + s0) * DH;
  const float* Kb = kbuf + (size_t)bh * S_LEN * DH;
  const float* Vb = vbuf + (size_t)bh * S_LEN * DH;

  // staging mapping: 256 threads cover 16 keys x 128 d in 16B pieces
  const int srow = tid >> 4;           // 0..15
  const int scol = (tid & 15) * 8;     // 0..120

  // preload Q A-fragments (16x128, 32 k-steps of 4)
  v2f qf[32];
  {
    const float* qrow = Qb + (size_t)ln * DH;
#pragma unroll
    for (int j = 0; j < 32; ++j)
      qf[j] = *(const v2f*)(qrow + j * 4 + 2 * hi);
  }

  v8f acc[8];                          // 16x128 output accumulator
#pragma unroll
  for (int t = 0; t < 8; ++t)
#pragma unroll
    for (int r = 0; r < 8; ++r) acc[t][r] = 0.0f;

  float rmax[8], rsum[8];
#pragma unroll
  for (int r = 0; r < 8; ++r) { rmax[r] = -1e30f; rsum[r] = 0.0f; }

  float* pw = &Ps[wave][0];

  // chunk stager: async copy valid keys, ds_store zeros for OOB keys
  auto stage = [&](int ci, int buf) {
    const int key = qb - 256 + ci * 16 + srow;
    float* kd = &Ks[buf][srow * 132 + scol];
    float* vd = &Vs[buf][srow * 132 + scol];
    if (key >= 0 && key < S_LEN) {
      const float* kg = Kb + (size_t)key * DH + scol;
      const float* vg = Vb + (size_t)key * DH + scol;
      cp16_g2l(kg, kd);
      cp16_g2l(kg + 4, kd + 4);
      cp16_g2l(vg, vd);
      cp16_g2l(vg + 4, vd + 4);
    } else {
      float4 z4; z4.x = z4.y = z4.z = z4.w = 0.0f;
      *(float4*)kd = z4; *(float4*)(kd + 4) = z4;
      *(float4*)vd = z4; *(float4*)(vd + 4) = z4;
    }
  };

  stage(0, 0);   // prologue

  for (int ci = 0; ci < 40; ++ci) {
    async_wait0();       // my copies for chunk ci landed
    __syncthreads();     // all waves' copies landed; prev reads of nxt done

    if (ci + 1 < 40) stage(ci + 1, (ci + 1) & 1);

    const int kbase = qb - 256 + ci * 16;
    const float* ks = Ks[ci & 1];
    const float* vs = Vs[ci & 1];

    // wave-uniform: does this chunk intersect my band window?
    if ((kbase + 15 >= s0 - WIN) && (kbase <= s0 + 15 + WIN)) {
      // S tile = Q (16x128) * K_chunk^T (128x16); 4 interleaved accumulators
      v8f sA[4];
#pragma unroll
      for (int t = 0; t < 4; ++t)
#pragma unroll
        for (int r = 0; r < 8; ++r) sA[t][r] = 0.0f;
#pragma unroll
      for (int j = 0; j < 32; ++j) {
        v2f b = *(const v2f*)&ks[ln * 132 + j * 4 + 2 * hi];
        sA[j & 3] = wmma4f32(qf[j], b, sA[j & 3]);
      }
      v8f sT = (sA[0] + sA[1]) + (sA[2] + sA[3]);

      // online softmax update (band mask -> -1e30)
      const int keyn = kbase + ln;     // this lane's key column
      float pv[8], scl[8];
#pragma unroll
      for (int r = 0; r < 8; ++r) {
        const int q = s0 + r + 8 * hi;
        float sv = sT[r];
        sv = (keyn >= q - WIN && keyn <= q + WIN) ? sv : -1e30f;
        float mx = sv;
        mx = fmaxf(mx, __shfl_xor(mx, 1));
        mx = fmaxf(mx, __shfl_xor(mx, 2));
        mx = fmaxf(mx, __shfl_xor(mx, 4));
        mx = fmaxf(mx, __shfl_xor(mx, 8));
        const float nm = fmaxf(rmax[r], mx);
        const float sc = __expf(rmax[r] - nm);
        const float p  = __expf(sv - nm);
        float ps = p;
        ps += __shfl_xor(ps, 1);
        ps += __shfl_xor(ps, 2);
        ps += __shfl_xor(ps, 4);
        ps += __shfl_xor(ps, 8);
        rsum[r] = rsum[r] * sc + ps;
        rmax[r] = nm;
        scl[r]  = sc;
        pv[r]   = p;
      }
#pragma unroll
      for (int t = 0; t < 8; ++t)
#pragma unroll
        for (int r = 0; r < 8; ++r) acc[t][r] *= scl[r];

      // bounce P (D-layout) through LDS to A-layout
#pragma unroll
      for (int r = 0; r < 8; ++r)
        pw[(r + 8 * hi) * 20 + ln] = pv[r];

      // O += P (16x16) * V_chunk (16x128)
#pragma unroll
      for (int j = 0; j < 4; ++j) {
        v2f a = *(const v2f*)&pw[ln * 20 + j * 4 + 2 * hi];
        const int kr = j * 4 + 2 * hi;
#pragma unroll
        for (int t = 0; t < 8; ++t) {
          v2f b;
          b.x = vs[kr * 132 + t * 16 + ln];
          b.y = vs[(kr + 1) * 132 + t * 16 + ln];
          acc[t] = wmma4f32(a, b, acc[t]);
        }
      }
    }
  }

  // normalize and store: out[s'][b'][h*128+d], out is (S, B, E)
  float rinv[8];
#pragma unroll
  for (int r = 0; r < 8; ++r) rinv[r] = 1.0f / rsum[r];

  const int h  = bh & 7;
  const int bp = bh >> 3;
#pragma unroll
  for (int t = 0; t < 8; ++t) {
#pragma unroll
    for (int r = 0; r < 8; ++r) {
      const int sp = s0 + r + 8 * hi;
      const int e  = h * DH + t * 16 + ln;
      out[(size_t)sp * (BSZ * EMB) + (size_t)bp * EMB + e] = acc[t][r] * rinv[r];
    }
  }
}

// ---------------------------------------------------------------------------
extern "C" void kernel_launch(void* const* d_in, const int* in_sizes, int n_in,
                              void* d_out, int out_size, void* d_ws, size_t ws_size,
                              hipStream_t stream) {
  (void)in_sizes; (void)n_in; (void)out_size; (void)ws_size;
  const float* val = (const float*)d_in[0];
  const float* Wq  = (const float*)d_in[1];
  const float* bq  = (const float*)d_in[2];
  const float* Wk  = (const float*)d_in[3];
  const float* bk  = (const float*)d_in[4];
  const float* Wv  = (const float*)d_in[5];
  const float* bv  = (const float*)d_in[6];
  float* out = (float*)d_out;

  const size_t perbuf = (size_t)BSZ * HEADS * S_LEN * DH;  // 8,388,608 floats
  float* qbuf = (float*)d_ws;
  float* kbuf = qbuf + perbuf;
  float* vbuf = kbuf + perbuf;

  dim3 gp(128, 16, 3), bpj(128);
  qkv_proj_kernel<<<gp, bpj, 0, stream>>>(val, Wq, bq, Wk, bk, Wv, bv,
                                          qbuf, kbuf, vbuf);
  dim3 ga(S_LEN / 128, BSZ * HEADS), ba(256);
  banded_attn_kernel<<<ga, ba, 0, stream>>>(qbuf, kbuf, vbuf, out);
}